// CondCapsuleLayer_75952201663093
// MI455X (gfx1250) — compile-verified
//
#include <hip/hip_runtime.h>
#include <hip/hip_bf16.h>
#include <math.h>

// Problem constants (from reference):
//   x:             [B=64,  R=1152, CI=64]           fp32
//   cond:          [B=64,  CI=64]   (UNUSED by reference math)
//   route_weights: [C=32,  R=1152, CI=64, O=32]     fp32
//   out:           [B=64,  C=32,   O=32]            fp32
#define B_   64
#define R_   1152
#define CI_  64
#define C_   32
#define O_   32

typedef __attribute__((ext_vector_type(2))) float v2f;
typedef __attribute__((ext_vector_type(8))) float v8f;

// ---------------------------------------------------------------------------
// CDNA5 async global->LDS staging (ASYNCcnt-tracked DMA), with sync fallback.
// Builtin signature (from hipcc diagnostic): takes addrspace(1)/(3) pointers
// to a 128-bit GCC-vector of int, plus imm offset and imm cpol.
// ---------------------------------------------------------------------------
#if defined(__has_builtin)
# if __has_builtin(__builtin_amdgcn_global_load_async_to_lds_b128)
#  define HAVE_ASYNC_LDS 1
# endif
#endif
#ifndef HAVE_ASYNC_LDS
# define HAVE_ASYNC_LDS 0
#endif

typedef int b128_t __attribute__((vector_size(16)));
typedef __attribute__((address_space(1))) b128_t* g_b128_p;
typedef __attribute__((address_space(3))) b128_t* l_b128_p;

__device__ __forceinline__ void cp16(const float* g, float* l) {
#if HAVE_ASYNC_LDS
    // uintptr_t round-trip: drops const, retargets addrspace (generic low
    // 32 bits == LDS offset per ISA aperture rules).
    __builtin_amdgcn_global_load_async_to_lds_b128(
        (g_b128_p)(uintptr_t)g, (l_b128_p)(uintptr_t)l, 0, 0);
#else
    *(float4*)l = *(const float4*)g;
#endif
}

#if HAVE_ASYNC_LDS
# if __has_builtin(__builtin_amdgcn_s_wait_asynccnt)
#  define WAIT_STAGE(n) __builtin_amdgcn_s_wait_asynccnt(n)
# else
#  define WAIT_STAGE(n) asm volatile("s_wait_asynccnt %0" :: "i"(n) : "memory")
# endif
#else
# define WAIT_STAGE(n) ((void)0)
#endif

// ---------------------------------------------------------------------------
// Phase 1: priors[c][b][r][o] = sum_i x[b,r,i] * W[c,r,i,o]
// Per workgroup: fixed c, RT consecutive r's. Per r: a 64x32x64 GEMM done as
// 4 (M=b) x 2 (N=o) WMMA tiles across 8 waves, K=64 in 16 steps of
// v_wmma_f32_16x16x4_f32 (exact fp32 — problem is HBM-bound, bf16 buys ~0).
// Double-buffered LDS: async-DMA r+1's operands while computing r.
// ---------------------------------------------------------------------------
#define RT 8                 // r rows per workgroup (1152 / 8 = 144 chunks)
#define XS_STRIDE 68         // 64 + 4 pad: conflict-free A reads, 16B aligned
#define WS_STRIDE 40         // 32 + 8 pad: conflict-free B reads, 16B aligned

__global__ __launch_bounds__(256, 2)
void capsule_priors_wmma(const float* __restrict__ x,
                         const float* __restrict__ w,
                         float* __restrict__ priors)
{
    __shared__ float xs[2][B_ * XS_STRIDE];   // x[:, r, :]    2 x 17.4 KB
    __shared__ float wl[2][CI_ * WS_STRIDE];  // W[c, r, :, :] 2 x 10.2 KB

    const int c    = blockIdx.y;
    const int r0   = blockIdx.x * RT;
    const int t    = threadIdx.x;
    const int wave = t >> 5;
    const int lane = t & 31;
    const int lm   = lane & 15;
    const int grp  = lane >> 4;        // half-wave: selects K pair per ISA layout
    const int mtile = wave & 3;        // which 16-row b tile
    const int ntile = wave >> 2;       // which 16-col o tile
    const int brow  = mtile * 16 + lm; // A-matrix M index (= batch b)
    const int ncol  = ntile * 16 + lm; // B/D N index (= out channel o)

    // Issue the 6 per-thread 16B chunks (4 for x, 2 for W) of row r into buf.
    auto issue_stage = [&](int r, int buf) {
        #pragma unroll
        for (int j = 0; j < 4; ++j) {
            int f  = t + 256 * j;          // 0..1023 float4's
            int bb = f >> 4;               // 16 float4 per b row (64 floats)
            int i4 = f & 15;
            cp16(&x[(size_t)(bb * R_ + r) * CI_ + i4 * 4],
                 &xs[buf][bb * XS_STRIDE + i4 * 4]);
        }
        #pragma unroll
        for (int j = 0; j < 2; ++j) {
            int f  = t + 256 * j;          // 0..511 float4's
            int k  = f >> 3;               // 8 float4 per i row (32 floats)
            int o4 = f & 7;
            cp16(&w[((size_t)(c * R_ + r) * CI_ + k) * O_ + o4 * 4],
                 &wl[buf][k * WS_STRIDE + o4 * 4]);
        }
    };

    issue_stage(r0, 0);                    // prologue: stage first row

    for (int rr = 0; rr < RT; ++rr) {
        const int r   = r0 + rr;
        const int cur = rr & 1;

        if (rr + 1 < RT) {
            issue_stage(r + 1, cur ^ 1);   // DMA next row while we compute
            WAIT_STAGE(6);                 // in-order: current row's 6 landed
        } else {
            WAIT_STAGE(0);
        }
        __syncthreads();                   // current buffer visible to all waves

        // ---- K loop: 16 x v_wmma_f32_16x16x4_f32, accumulate in VGPRs ----
        v8f acc = {0.f, 0.f, 0.f, 0.f, 0.f, 0.f, 0.f, 0.f};
        #pragma unroll
        for (int k4 = 0; k4 < 16; ++k4) {
            const int kb = k4 * 4 + grp * 2;   // lanes 0-15: K=0,1; 16-31: K=2,3
            v2f av, bv;
            av[0] = xs[cur][brow * XS_STRIDE + kb + 0];
            av[1] = xs[cur][brow * XS_STRIDE + kb + 1];
            bv[0] = wl[cur][(kb + 0) * WS_STRIDE + ncol];
            bv[1] = wl[cur][(kb + 1) * WS_STRIDE + ncol];
            acc = __builtin_amdgcn_wmma_f32_16x16x4_f32(
                false, av, false, bv, (short)0, acc, false, false);
        }

        // ---- store D tile: priors[c][m][r][n]; layout chosen so phase 2
        //      reads each (c,b) tile fully coalesced ----
        #pragma unroll
        for (int vv = 0; vv < 8; ++vv) {
            int m = mtile * 16 + grp * 8 + vv; // C/D layout: VGPR v -> M=v(+8)
            priors[(((size_t)c * B_ + m) * R_ + r) * O_ + ncol] = acc[vv];
        }
        __syncthreads();   // all reads of buf `cur` done before it is re-DMA'd
    }
}

// ---------------------------------------------------------------------------
// Phase 2: dynamic routing, one workgroup per (c,b). 12 waves; wave w owns
// routes r in [96w, 96w+96), lane = output channel o. The 1152x32 priors tile
// (144 KB) lives in registers (96 f32/lane); logits/probs in LDS; all 3
// iterations run on-chip so priors come from HBM exactly once.
// ---------------------------------------------------------------------------
#define THREADS2 384
#define WAVES2   12
#define RPW      96            // routes per wave

__global__ __launch_bounds__(THREADS2, 1)
void capsule_routing(const float* __restrict__ priors, float* __restrict__ out)
{
    __shared__ float lg[R_];              // logits
    __shared__ float pb[R_];              // exp(logit - max)
    __shared__ float red[WAVES2];         // cross-wave scalar reduce
    __shared__ float pred[WAVES2][O_];    // cross-wave vec32 reduce
    __shared__ float svec[O_];            // squashed output v
    __shared__ float scal[2];             // softmax max, 1/denom

    const int b    = blockIdx.x;
    const int c    = blockIdx.y;
    const int t    = threadIdx.x;
    const int wave = t >> 5;
    const int lane = t & 31;

    // Coalesced register load of this (c,b) priors tile.
    const float* base = priors + ((size_t)c * B_ + b) * R_ * O_;
    float pri[RPW];
    #pragma unroll
    for (int j = 0; j < RPW; ++j)
        pri[j] = base[(size_t)(wave * RPW + j) * O_ + lane];

    for (int r = t; r < R_; r += THREADS2) lg[r] = 0.f;
    __syncthreads();

    float vout = 0.f;

    for (int iter = 0; iter < 3; ++iter) {
        // ---- stable softmax over R: block max ----
        float lmax = -3.4e38f;
        for (int r = t; r < R_; r += THREADS2) lmax = fmaxf(lmax, lg[r]);
        #pragma unroll
        for (int sh = 16; sh > 0; sh >>= 1)
            lmax = fmaxf(lmax, __shfl_xor(lmax, sh, 32));
        if (lane == 0) red[wave] = lmax;
        __syncthreads();
        if (t == 0) {
            float m = red[0];
            for (int i = 1; i < WAVES2; ++i) m = fmaxf(m, red[i]);
            scal[0] = m;
        }
        __syncthreads();
        const float M = scal[0];

        // ---- exp + block sum ----
        float lsum = 0.f;
        for (int r = t; r < R_; r += THREADS2) {
            float e = __expf(lg[r] - M);
            pb[r] = e;
            lsum += e;
        }
        #pragma unroll
        for (int sh = 16; sh > 0; sh >>= 1) lsum += __shfl_xor(lsum, sh, 32);
        if (lane == 0) red[wave] = lsum;
        __syncthreads();
        if (t == 0) {
            float d = 0.f;
            for (int i = 0; i < WAVES2; ++i) d += red[i];
            scal[1] = 1.f / d;
        }
        __syncthreads();
        const float invden = scal[1];

        // ---- s[o] = sum_r probs[r] * priors[r,o]  (o = lane) ----
        float accp = 0.f;
        #pragma unroll
        for (int j = 0; j < RPW; ++j)
            accp += pb[wave * RPW + j] * pri[j];
        pred[wave][lane] = accp * invden;
        __syncthreads();

        if (wave == 0) {
            float s = 0.f;
            #pragma unroll
            for (int i = 0; i < WAVES2; ++i) s += pred[i][lane];
            // squash: v = s * sqrt(sq) / (1 + sq),  sq = |s|^2
            float sq = s * s;
            #pragma unroll
            for (int sh = 16; sh > 0; sh >>= 1) sq += __shfl_xor(sq, sh, 32);
            float v = s * (sqrtf(sq) / (1.f + sq));
            svec[lane] = v;
            vout = v;
        }
        __syncthreads();

        if (iter < 2) {
            // logits[r] += dot(priors[r,:], v) — wave butterfly over o=lanes.
            const float vl = svec[lane];
            #pragma unroll
            for (int j = 0; j < RPW; ++j) {
                float d = pri[j] * vl;
                #pragma unroll
                for (int sh = 16; sh > 0; sh >>= 1) d += __shfl_xor(d, sh, 32);
                if (lane == 0) lg[wave * RPW + j] += d;
            }
            __syncthreads();
        }
    }

    // out is [B, C, O] (reference transposes [C,B,O] -> [B,C,O])
    if (wave == 0) out[((size_t)b * C_ + c) * O_ + lane] = vout;
}

// ---------------------------------------------------------------------------
extern "C" void kernel_launch(void* const* d_in, const int* in_sizes, int n_in,
                              void* d_out, int out_size, void* d_ws, size_t ws_size,
                              hipStream_t stream)
{
    const float* x = (const float*)d_in[0];
    // d_in[1] = cond — unused by the reference computation.
    const float* w = (const float*)d_in[2];
    float* priors  = (float*)d_ws;   // needs C_*B_*R_*O_*4 = 301,989,888 bytes
    float* out     = (float*)d_out;

    dim3 g1(R_ / RT, C_);            // 144 x 32 workgroups
    capsule_priors_wmma<<<g1, 256, 0, stream>>>(x, w, priors);

    dim3 g2(B_, C_);                 // one workgroup per (c,b)
    capsule_routing<<<g2, THREADS2, 0, stream>>>(priors, out);
}